// SubstructureAttention_71734543777917
// MI455X (gfx1250) — compile-verified
//
#include <hip/hip_runtime.h>
#include <math.h>

typedef __attribute__((ext_vector_type(2))) float v2f;
typedef __attribute__((ext_vector_type(4))) float v4f;
typedef __attribute__((ext_vector_type(8))) float v8f;

#define HD 256      // hidden dim
#define NA 64       // atoms per graph
#define NG 64       // graphs
#define MTOT (NG * NA)   // 4096 rows per GEMM

// ---------------------------------------------------------------------------
// Kernel 1: fp32 WMMA GEMM.  S_half = A_half @ W1_half   (M=4096, N=256, K=256)
//   blockIdx.z = 0 : s1 = h1 @ W1[0:256,:]
//   blockIdx.z = 1 : s2 = h2 @ W1[256:512,:]
// One wave computes a 16x32 strip: 2 accumulators sharing the A fetch.
// V_WMMA_F32_16X16X4_F32 layouts (ISA 7.12.2):
//   A 16x4 : lanes 0-15 -> K=0(v0),K=1(v1) ; lanes 16-31 -> K=2(v0),K=3(v1)
//   B 4x16 : lanes 0-15 -> K=0(v0),K=1(v1) ; lanes 16-31 -> K=2(v0),K=3(v1)
//   C 16x16: VGPR r -> row r (lanes 0-15) / row r+8 (lanes 16-31)
// ---------------------------------------------------------------------------
__global__ __launch_bounds__(128)
void substructure_gemm_wmma(const float* __restrict__ h1,
                            const float* __restrict__ h2,
                            const float* __restrict__ W1,
                            float* __restrict__ s1,
                            float* __restrict__ s2) {
  const int lane   = threadIdx.x & 31;
  const int wave   = threadIdx.x >> 5;
  const int laneLo = lane & 15;
  const int kHi    = (lane >> 4) * 2;          // 0 for lanes 0-15, 2 for 16-31

  const int mt   = blockIdx.x;                 // 0..255 : 16-row tile
  const int nt2  = blockIdx.y * 4 + wave;      // 0..7   : 32-col strip
  const int half = blockIdx.z;

  const float* __restrict__ A = half ? h2 : h1;
  const float* __restrict__ W = W1 + (size_t)half * HD * HD;  // row offset 256
  float* __restrict__ S       = half ? s2 : s1;

  const int m  = mt * 16 + laneLo;
  const int n0 = nt2 * 32 + laneLo;
  const int n1 = n0 + 16;

  const float* __restrict__ Arow = A + (size_t)m * HD;

  v8f c0 = {};
  v8f c1 = {};

#pragma unroll 4
  for (int k0 = 0; k0 < HD; k0 += 4) {
    // A fragment: contiguous pair of K values for this lane -> one b64 load
    v2f a = *(const v2f*)(Arow + k0 + kHi);
    // B fragments: rows (k0+kHi) and (k0+kHi+1) of W, column n
    const float* __restrict__ Wk = W + (size_t)(k0 + kHi) * HD;
    v2f b0 = { Wk[n0], Wk[HD + n0] };
    v2f b1 = { Wk[n1], Wk[HD + n1] };
    c0 = __builtin_amdgcn_wmma_f32_16x16x4_f32(false, a, false, b0,
                                               (short)0, c0, false, false);
    c1 = __builtin_amdgcn_wmma_f32_16x16x4_f32(false, a, false, b1,
                                               (short)0, c1, false, false);
  }

  const int rowBase = mt * 16 + ((lane >> 4) ? 8 : 0);
#pragma unroll
  for (int r = 0; r < 8; ++r) {
    S[(size_t)(rowBase + r) * HD + n0] = c0[r];
    S[(size_t)(rowBase + r) * HD + n1] = c1[r];
  }
}

// ---------------------------------------------------------------------------
// Kernel 2: per-graph fused  tanh-score -> softmax(4096) -> row/col marginals.
// One block (256 thr = 8 waves) per graph. s2 tile lives in padded LDS;
// the [64,64,256] "hidden" tensor is never materialized.
// Thread t owns (i = t>>2, j in [(t&3)*16, +16)).
// b2 is a uniform additive constant -> cancels in softmax, omitted.
// ---------------------------------------------------------------------------
#define S2_STRIDE 257   // 16*257 % 64 == 16 -> 4 distinct j's hit distinct banks
#define SC_STRIDE 65

__global__ __launch_bounds__(256)
void substructure_score_softmax(const float* __restrict__ s1,
                                const float* __restrict__ s2,
                                const float* __restrict__ b1,
                                const float* __restrict__ w2,
                                float* __restrict__ out) {
  __shared__ float s2s[NA * S2_STRIDE];   // 65792 B
  __shared__ float sc[NA * SC_STRIDE];    // 16640 B
  __shared__ float cb1[HD];
  __shared__ float cw2[HD];
  __shared__ float red[8];
  __shared__ float gbuf[2];

  const int t = threadIdx.x;
  const int b = blockIdx.x;

  // broadcast constants (blockDim == HD)
  cb1[t] = b1[t];
  cw2[t] = w2[t];

  // stage this graph's s2 tile into LDS (vector global loads, scalar LDS
  // stores because of the odd padding stride)
  const float* __restrict__ S2 = s2 + (size_t)b * NA * HD;
  for (int e = t; e < NA * HD / 4; e += 256) {
    v4f v = *(const v4f*)(S2 + e * 4);
    const int j = (e * 4) >> 8;
    const int h = (e * 4) & (HD - 1);
    float* dst = &s2s[j * S2_STRIDE + h];
    dst[0] = v.x; dst[1] = v.y; dst[2] = v.z; dst[3] = v.w;
  }
  __syncthreads();

  const int i  = t >> 2;
  const int j0 = (t & 3) * 16;
  const float* __restrict__ S1row = s1 + ((size_t)b * NA + i) * HD;

  float acc[16];
#pragma unroll
  for (int jj = 0; jj < 16; ++jj) acc[jj] = 0.0f;

  for (int h0 = 0; h0 < HD; h0 += 8) {
    float a[8], bb[8], ww[8];
#pragma unroll
    for (int d = 0; d < 8; ++d) {
      a[d]  = S1row[h0 + d];     // vectorizes to 2x b128
      bb[d] = cb1[h0 + d];
      ww[d] = cw2[h0 + d];
    }
#pragma unroll
    for (int jj = 0; jj < 16; ++jj) {
      const float* __restrict__ srow = &s2s[(j0 + jj) * S2_STRIDE + h0];
      float p = 0.0f;
#pragma unroll
      for (int d = 0; d < 8; ++d)
        p += tanhf(a[d] + srow[d] + bb[d]) * ww[d];
      acc[jj] += p;
    }
  }

  // ---- softmax over all 4096 scores of this graph ----
  float lmax = acc[0];
#pragma unroll
  for (int jj = 1; jj < 16; ++jj) lmax = fmaxf(lmax, acc[jj]);
#pragma unroll
  for (int off = 16; off > 0; off >>= 1)
    lmax = fmaxf(lmax, __shfl_xor(lmax, off, 32));
  if ((t & 31) == 0) red[t >> 5] = lmax;
  __syncthreads();
  if (t == 0) {
    float m = red[0];
#pragma unroll
    for (int w = 1; w < 8; ++w) m = fmaxf(m, red[w]);
    gbuf[0] = m;
  }
  __syncthreads();
  const float gmax = gbuf[0];

  float e[16];
  float lsum = 0.0f;
#pragma unroll
  for (int jj = 0; jj < 16; ++jj) {
    e[jj] = __expf(acc[jj] - gmax);
    lsum += e[jj];
  }
#pragma unroll
  for (int off = 16; off > 0; off >>= 1)
    lsum += __shfl_xor(lsum, off, 32);
  if ((t & 31) == 0) red[t >> 5] = lsum;
  __syncthreads();
  if (t == 0) {
    float s = 0.0f;
#pragma unroll
    for (int w = 0; w < 8; ++w) s += red[w];
    gbuf[1] = 1.0f / s;
  }
  __syncthreads();
  const float inv = gbuf[1];

  // normalized probabilities into padded LDS tile
#pragma unroll
  for (int jj = 0; jj < 16; ++jj)
    sc[i * SC_STRIDE + j0 + jj] = e[jj] * inv;
  __syncthreads();

  // exact marginals: attention_1 = row sums, attention_2 = col sums
  if (t < NA) {
    float rs = 0.0f, cs = 0.0f;
#pragma unroll 8
    for (int x = 0; x < NA; ++x) {
      rs += sc[t * SC_STRIDE + x];   // row t over j
      cs += sc[x * SC_STRIDE + t];   // col t over i
    }
    out[(size_t)b * NA + t]                  = rs;   // attention_1
    out[(size_t)MTOT + (size_t)b * NA + t]   = cs;   // attention_2
  }
}

// ---------------------------------------------------------------------------
extern "C" void kernel_launch(void* const* d_in, const int* in_sizes, int n_in,
                              void* d_out, int out_size, void* d_ws, size_t ws_size,
                              hipStream_t stream) {
  (void)in_sizes; (void)n_in; (void)out_size; (void)ws_size;
  const float* h1 = (const float*)d_in[0];   // [4096,256]
  const float* h2 = (const float*)d_in[1];   // [4096,256]
  // d_in[2], d_in[3]: int64 batch ids (sorted equal-size -> implicit reshape)
  // d_in[4]: n_graphs scalar (fixed 64); d_in[8]: b2 (cancels in softmax)
  const float* W1 = (const float*)d_in[5];   // [512,256]
  const float* b1 = (const float*)d_in[6];   // [256]
  const float* W2 = (const float*)d_in[7];   // [256,1]
  float* out = (float*)d_out;                // [8192] = att1 ++ att2

  float* s1 = (float*)d_ws;                  // [4096,256]
  float* s2 = s1 + (size_t)MTOT * HD;        // [4096,256]  (8 MB total)

  dim3 g1(MTOT / 16, 2, 2);                  // (256,2,2)
  substructure_gemm_wmma<<<g1, 128, 0, stream>>>(h1, h2, W1, s1, s2);

  substructure_score_softmax<<<NG, 256, 0, stream>>>(s1, s2, b1, W2, out);
}